// AdditiveAttention_7533372637540
// MI455X (gfx1250) — compile-verified
//
#include <hip/hip_runtime.h>
#include <hip/hip_bf16.h>

// Problem constants (from reference): B=8, Lq=256, Lk=512, D=256 (=dq=dv), H=64
#define BB 8
#define LQ 256
#define LK 512
#define DD 256
#define HH 64
#define DV 256
#define NEG_INF_MASK (-1e6f)

typedef __attribute__((ext_vector_type(2))) float v2f;
typedef __attribute__((ext_vector_type(8))) float v8f;

__device__ __forceinline__ v8f wmma_f32_k4(v2f a, v2f b, v8f c) {
  // D(16x16,f32) = A(16x4,f32) * B(4x16,f32) + C
  return __builtin_amdgcn_wmma_f32_16x16x4_f32(
      /*neg_a=*/false, a, /*neg_b=*/false, b,
      /*c_mod=*/(short)0, c, /*reuse_a=*/false, /*reuse_b=*/false);
}

__device__ __forceinline__ float fast_tanh(float x) {
#if __has_builtin(__builtin_amdgcn_tanhf)
  return __builtin_amdgcn_tanhf(x);   // v_tanh_f32 (TRANS pipe, co-issues with VALU FMA)
#else
  return tanhf(x);
#endif
}

// Async global->LDS b128 copy (ASYNCcnt path). LDS operand = low 32 bits of the
// generic shared pointer (flat->LDS uses addr[31:0] as LDS offset, ISA 10.2).
__device__ __forceinline__ void async_copy_b128(const float* gsrc, float* lds_dst) {
  unsigned laddr = (unsigned)(unsigned long long)lds_dst;
  asm volatile("global_load_async_to_lds_b128 %0, %1, off"
               :: "v"(laddr), "v"(gsrc)
               : "memory");
}

__device__ __forceinline__ void wait_async0() {
#if __has_builtin(__builtin_amdgcn_s_wait_asynccnt)
  __builtin_amdgcn_s_wait_asynccnt(0);
#else
  asm volatile("s_wait_asynccnt 0x0" ::: "memory");
#endif
}

// ---------------------------------------------------------------------------
// Kernel 1/2: Y[b,l,h] = sum_d X[b,l,d] * W[h,d]      (X:[B,L,D], W:[H,D])
// One wave computes one 16(M=L) x 16(N=H) f32 tile with 64 chained
// v_wmma_f32_16x16x4_f32 steps. 4 waves/block cover H=64.
// ---------------------------------------------------------------------------
__global__ void __launch_bounds__(128)
proj_kernel(const float* __restrict__ X, const float* __restrict__ W,
            float* __restrict__ Y, int L) {
  const int lane = threadIdx.x & 31;
  const int wv   = threadIdx.x >> 5;          // 0..3 -> H tile
  const int tile = blockIdx.x;                // b * (L/16) + mtile
  const int b    = tile / (L >> 4);
  const int m0   = (tile % (L >> 4)) << 4;
  const int h0   = wv << 4;

  const int lm  = lane & 15;
  const int khi = (lane >> 4) << 1;           // 0 or 2

  const float* arow = X + ((size_t)(b * L + m0 + lm)) * DD + khi;
  const float* brow = W + ((size_t)(h0 + lm)) * DD + khi;

  v8f c = {};
#pragma unroll 8
  for (int kk = 0; kk < DD; kk += 4) {
    v2f a  = *(const v2f*)(arow + kk);        // K{khi, khi+1}
    v2f bb = *(const v2f*)(brow + kk);
    c = wmma_f32_k4(a, bb, c);
  }

  const int mb = (lane >> 4) << 3;            // 0 or 8
  float* out = Y + ((size_t)(b * L + m0 + mb)) * HH + h0 + lm;
#pragma unroll
  for (int r = 0; r < 8; ++r) out[(size_t)r * HH] = c[r];
}

// ---------------------------------------------------------------------------
// Kernel 3: scores + masked softmax, fused. One 256-thread block per (b,q) row.
// score[k] = sum_h w[h]*tanh(qp[h]+kp[k,h]); each thread handles 2 k's.
// 64 tanh per k fully unrolled: TRANS (v_tanh) co-executes with VALU FMAs.
// ---------------------------------------------------------------------------
__device__ __forceinline__ float wave_max(float v) {
#pragma unroll
  for (int o = 16; o > 0; o >>= 1) v = fmaxf(v, __shfl_xor(v, o, 32));
  return v;
}
__device__ __forceinline__ float wave_sum(float v) {
#pragma unroll
  for (int o = 16; o > 0; o >>= 1) v += __shfl_xor(v, o, 32);
  return v;
}

__global__ void __launch_bounds__(256)
scores_softmax_kernel(const float* __restrict__ qp, const float* __restrict__ kp,
                      const float* __restrict__ wv, const int* __restrict__ valid_lens,
                      float* __restrict__ attn) {
  const int row = blockIdx.x;                 // b*LQ + qi
  const int b   = row / LQ;
  const int tid = threadIdx.x;

  __shared__ float sh_q[HH];
  __shared__ float sh_w[HH];
  __shared__ float red[8];

  if (tid < HH) {
    sh_q[tid] = qp[(size_t)row * HH + tid];
    sh_w[tid] = wv[tid];
  }
  __syncthreads();

  const int valid = valid_lens[b];
  float s[2];
#pragma unroll
  for (int i = 0; i < 2; ++i) {
    const int k = tid + (i << 8);
    const float4* k4 = (const float4*)(kp + ((size_t)(b * LK + k)) * HH);
    float acc = 0.f;
#pragma unroll
    for (int h4 = 0; h4 < HH / 4; ++h4) {
      float4 kvv = k4[h4];
      const int h = h4 << 2;
      acc += sh_w[h + 0] * fast_tanh(sh_q[h + 0] + kvv.x);
      acc += sh_w[h + 1] * fast_tanh(sh_q[h + 1] + kvv.y);
      acc += sh_w[h + 2] * fast_tanh(sh_q[h + 2] + kvv.z);
      acc += sh_w[h + 3] * fast_tanh(sh_q[h + 3] + kvv.w);
    }
    s[i] = (k < valid) ? acc : NEG_INF_MASK;
  }

  const int wid = tid >> 5, ln = tid & 31;
  float m = wave_max(fmaxf(s[0], s[1]));
  if (ln == 0) red[wid] = m;
  __syncthreads();
  if (wid == 0) {
    float t = (ln < 8) ? red[ln] : -3.4e38f;
    t = wave_max(t);
    if (ln == 0) red[0] = t;
  }
  __syncthreads();
  const float gmax = red[0];
  __syncthreads();

  const float e0 = __expf(s[0] - gmax);
  const float e1 = __expf(s[1] - gmax);

  float sm = wave_sum(e0 + e1);
  if (ln == 0) red[wid] = sm;
  __syncthreads();
  if (wid == 0) {
    float t = (ln < 8) ? red[ln] : 0.f;
    t = wave_sum(t);
    if (ln == 0) red[0] = t;
  }
  __syncthreads();
  const float inv = 1.0f / red[0];

  float* arow = attn + (size_t)row * LK;
  arow[tid]       = e0 * inv;
  arow[tid + 256] = e1 * inv;
}

// ---------------------------------------------------------------------------
// Kernel 4: out[b,q,v] = sum_k attn[b,q,k] * values[b,k,v]
// Block (128 thr = 4 waves) computes a 16(M) x 64(N) output tile.
// values K-slabs (64x64) are double-buffered in LDS via async global->LDS
// b128 copies (ASYNCcnt); all 4 waves share each slab. A rows stream from
// global as contiguous b64 loads. 16 WMMA k4 steps per 64-wide chunk.
// ---------------------------------------------------------------------------
#define KC 64
#define VSTRIDE 68   // floats; 68*4=272 bytes: 16B-aligned rows, bank-skewed

__global__ void __launch_bounds__(128)
pv_kernel(const float* __restrict__ attn, const float* __restrict__ values,
          float* __restrict__ out) {
  __shared__ float lds_v[2][KC * VSTRIDE];    // 2 x 17 KB

  const int tid  = threadIdx.x;
  const int lane = tid & 31;
  const int wv   = tid >> 5;                  // 0..3 -> 16-wide N subtile

  const int b   = blockIdx.x >> 6;            // 64 blocks per batch
  const int rem = blockIdx.x & 63;
  const int m0  = (rem >> 2) << 4;            // 16 M tiles
  const int n0b = (rem & 3) << 6;             // 4 N slabs of 64

  const int lm  = lane & 15;
  const int khi = (lane >> 4) << 1;

  const float* arow  = attn   + ((size_t)(b * LQ + m0 + lm)) * LK + khi;
  const float* vbase = values + (size_t)b * LK * DV + n0b;

  // prefetch chunk 0: 64 rows x 64 cols, 8 b128 transfers per thread
#pragma unroll
  for (int i = 0; i < 8; ++i) {
    const int idx = tid + (i << 7);
    const int row = idx >> 4, col = (idx & 15) << 2;
    async_copy_b128(vbase + (size_t)row * DV + col,
                    &lds_v[0][row * VSTRIDE + col]);
  }

  v8f c = {};
  for (int ch = 0; ch < LK / KC; ++ch) {
    wait_async0();            // this wave's pending slab writes done
    __syncthreads();          // every wave's writes done -> slab readable

    if (ch + 1 < LK / KC) {   // prefetch next slab into the other buffer
      const float* src = vbase + (size_t)(ch + 1) * KC * DV;
      float* dst = lds_v[(ch + 1) & 1];
#pragma unroll
      for (int i = 0; i < 8; ++i) {
        const int idx = tid + (i << 7);
        const int row = idx >> 4, col = (idx & 15) << 2;
        async_copy_b128(src + (size_t)row * DV + col, dst + row * VSTRIDE + col);
      }
    }

    const float* bb_base = &lds_v[ch & 1][wv * 16 + lm];
    const float* ar = arow + ch * KC;
#pragma unroll 4
    for (int kk = 0; kk < KC; kk += 4) {
      v2f a = *(const v2f*)(ar + kk);
      const int kb = kk + khi;
      v2f bb;
      bb.x = bb_base[(kb + 0) * VSTRIDE];
      bb.y = bb_base[(kb + 1) * VSTRIDE];
      c = wmma_f32_k4(a, bb, c);
    }
    __syncthreads();          // done reading this buffer before it's refilled
  }

  const int mb = (lane >> 4) << 3;
  float* orow = out + ((size_t)(b * LQ + m0 + mb)) * DV + n0b + (wv << 4) + lm;
#pragma unroll
  for (int r = 0; r < 8; ++r) orow[(size_t)r * DV] = c[r];
}

// ---------------------------------------------------------------------------
extern "C" void kernel_launch(void* const* d_in, const int* in_sizes, int n_in,
                              void* d_out, int out_size, void* d_ws, size_t ws_size,
                              hipStream_t stream) {
  (void)in_sizes; (void)n_in; (void)out_size; (void)ws_size;

  const float* queries    = (const float*)d_in[0];   // [8,256,256]
  const float* keys       = (const float*)d_in[1];   // [8,512,256]
  const float* values     = (const float*)d_in[2];   // [8,512,256]
  const int*   valid_lens = (const int*)  d_in[3];   // [8]
  const float* W_q        = (const float*)d_in[4];   // [64,256]
  const float* W_k        = (const float*)d_in[5];   // [64,256]
  const float* w_v        = (const float*)d_in[6];   // [64]
  float* out = (float*)d_out;                        // [8,256,256]

  float* qp   = (float*)d_ws;                        // [8,256,64]  512 KB
  float* kp   = qp + (size_t)BB * LQ * HH;           // [8,512,64]  1 MB
  float* attn = kp + (size_t)BB * LK * HH;           // [8,256,512] 4 MB

  // q / k projections (WMMA f32 16x16x4)
  proj_kernel<<<BB * (LQ / 16), 128, 0, stream>>>(queries, W_q, qp, LQ);
  proj_kernel<<<BB * (LK / 16), 128, 0, stream>>>(keys,    W_k, kp, LK);

  // fused additive scores + masked softmax (tanh on TRANS pipe)
  scores_softmax_kernel<<<BB * LQ, 256, 0, stream>>>(qp, kp, w_v, valid_lens, attn);

  // attn @ values (WMMA + async global->LDS double-buffered B slabs)
  pv_kernel<<<BB * (LQ / 16) * (DV / 64), 128, 0, stream>>>(attn, values, out);
}